// CTCLoss_49366354100510
// MI455X (gfx1250) — compile-verified
//
#include <hip/hip_runtime.h>
#include <stdint.h>

#define B_     64
#define T_     2048
#define V_     128
#define S_     128
#define L_     257        // 2*S+1 extended states
#define BLANK_ 127
#define NEGF   (-1e30f)
#define SPL    9          // states per lane: 32*9 = 288 >= 257
#define RPB    16         // timesteps (rows) per chunk
#define NBLK   (T_ / RPB) // 128 chunks
#define NBUF   3          // LDS row-chunk buffers

typedef float v2f __attribute__((ext_vector_type(2)));
typedef float v8f __attribute__((ext_vector_type(8)));

#if __has_builtin(__builtin_amdgcn_wmma_f32_16x16x4_f32)
#define USE_WMMA 1
#endif

typedef __attribute__((address_space(3))) const void* lds_cptr_t;

__device__ __forceinline__ uint32_t lds_off_u32(const void* p) {
  // generic -> LDS addrspace, take the 32-bit LDS byte offset
  return (uint32_t)(unsigned long long)(lds_cptr_t)p;
}

// logaddexp, safe for the -1e30 sentinel (finite, like the reference)
__device__ __forceinline__ float lae2(float x, float y) {
  float m = fmaxf(x, y);
  float d = fminf(x, y) - m;            // <= 0
  return m + __logf(1.0f + __expf(d));
}

__global__ __launch_bounds__(32, 1)
void ctc_alpha_kernel(const float* __restrict__ outputs,
                      const int*  __restrict__ targets,
                      float* __restrict__ loss_ws) {
  __shared__ __align__(16) float rows[NBUF][RPB][V_]; // 24 KB streamed logit rows
  __shared__ float lse_s[RPB];
  __shared__ float rmax_s[RPB];

  const int b    = blockIdx.x;
  const int lane = threadIdx.x;          // wave32: one wave per batch element
  const int* tb  = targets + b * S_;

  // ---- per-lane extended-label indices and skip flags (states lane*9 .. lane*9+8)
  int extv[SPL];
  int skpv[SPL];
#pragma unroll
  for (int k = 0; k < SPL; ++k) {
    int s = lane * SPL + k;
    int e = BLANK_, sk = 0;
    if (s < L_ && (s & 1)) {
      int i = (s - 1) >> 1;
      e = tb[i];
      if (s >= 3) sk = (e != tb[i - 1]) ? 1 : 0;
    }
    extv[k] = e;
    skpv[k] = sk;
  }

  // ---- alpha registers; virtual "pre" state: state 0 = 0, rest = NEG
  float a[SPL];
#pragma unroll
  for (int k = 0; k < SPL; ++k) a[k] = NEGF;
  if (lane == 0) a[0] = 0.0f;

  const uint32_t lane16 = (uint32_t)lane * 16u; // byte offset of this lane in a 512B row

  // one async b128 per row: 32 lanes x 16B = full 512B row, global -> LDS, no VGPR round trip
  auto issue_block = [&](int blk) {
    const int slot = blk % NBUF;
    const float* gbase = outputs + ((size_t)b * T_ + (size_t)blk * RPB) * V_;
#pragma unroll
    for (int r = 0; r < RPB; ++r) {
      uint32_t ldsa  = lds_off_u32(&rows[slot][r][0]) + lane16;
      uint64_t sbase = (uint64_t)(uintptr_t)(gbase + r * V_);
      asm volatile("global_load_async_to_lds_b128 %0, %1, %2"
                   :: "v"(ldsa), "v"(lane16), "s"(sbase)
                   : "memory");
    }
  };

  issue_block(0);
  issue_block(1);
  issue_block(2);

  for (int blk = 0; blk < NBLK; ++blk) {
    const int slot = blk % NBUF;
    asm volatile("s_wait_asynccnt 0" ::: "memory"); // chunk `blk` resident in LDS

    // ================= Phase B: logsumexp for 16 rows =================
    const int myrow = lane & 15;   // two lanes per row
    const int half  = lane >> 4;   // each handles 64 elements
    const float* rp = &rows[slot][myrow][half * 64];
    float m = NEGF;
#pragma unroll
    for (int i = 0; i < 16; ++i) {
      float4 v = ((const float4*)rp)[i];
      m = fmaxf(m, fmaxf(fmaxf(v.x, v.y), fmaxf(v.z, v.w)));
    }
    m = fmaxf(m, __shfl_xor(m, 16, 32)); // full row max in both lanes
    if (lane < 16) rmax_s[lane] = m;
    __builtin_amdgcn_wave_barrier();

#ifdef USE_WMMA
    // Row sums of exp() via f32 WMMA against an all-ones B (16x16x4, K-chunked 32x).
    // A layout (16x4 f32): lanes 0-15 hold K=0,1; lanes 16-31 hold K=2,3; M = lane&15.
    v8f acc = {0.f, 0.f, 0.f, 0.f, 0.f, 0.f, 0.f, 0.f};
    const v2f bones = {1.0f, 1.0f};
#pragma unroll
    for (int c = 0; c < 32; ++c) {
      const float* p = &rows[slot][myrow][4 * c + half * 2];
      v2f av;
      av.x = __expf(p[0] - m);
      av.y = __expf(p[1] - m);
      acc = __builtin_amdgcn_wmma_f32_16x16x4_f32(false, av, false, bones,
                                                  (short)0, acc, false, false);
    }
    // D layout: vgpr r -> row r (lanes 0-15) / row r+8 (lanes 16-31); all N equal.
    if (lane == 0 || lane == 16) {
#pragma unroll
      for (int r = 0; r < 8; ++r) {
        int row = half * 8 + r;
        lse_s[row] = rmax_s[row] + __logf(acc[r]);
      }
    }
#else
    float ss = 0.0f;
#pragma unroll
    for (int i = 0; i < 16; ++i) {
      float4 v = ((const float4*)rp)[i];
      ss += __expf(v.x - m) + __expf(v.y - m) + __expf(v.z - m) + __expf(v.w - m);
    }
    ss += __shfl_xor(ss, 16, 32);
    if (lane < 16) lse_s[lane] = m + __logf(ss);
#endif
    __builtin_amdgcn_wave_barrier();

    // ================= Phase C: 16 sequential alpha steps =================
    for (int tl = 0; tl < RPB; ++tl) {
      const float  lse = lse_s[tl];
      const float* row = &rows[slot][tl][0];
      // halo: last two states of the previous lane
      float h1 = __shfl_up(a[SPL - 1], 1, 32);
      float h2 = __shfl_up(a[SPL - 2], 1, 32);
      if (lane == 0) { h1 = NEGF; h2 = NEGF; }
      float an[SPL];
#pragma unroll
      for (int k = 0; k < SPL; ++k) {
        float p1 = (k == 0) ? h1 : a[k - 1];
        float p2 = (k == 0) ? h2 : ((k == 1) ? h1 : a[k - 2]);
        float lp = row[extv[k]] - lse;                 // LDS gather of label logit
        float s2 = skpv[k] ? p2 : NEGF;
        an[k] = lp + lae2(lae2(a[k], p1), s2);
      }
#pragma unroll
      for (int k = 0; k < SPL; ++k) a[k] = an[k];
    }

    // ================= prefetch: reuse the slot we just drained =================
    asm volatile("s_wait_dscnt 0" ::: "memory"); // all LDS reads of this slot retired
    if (blk + NBUF < NBLK) issue_block(blk + NBUF);
  }

  // ---- finish: loss_b = -logaddexp(alpha[L-1], alpha[L-2]);  state 255/256 live in lane 28
  float v255 = __shfl(a[3], 28, 32);
  float v256 = __shfl(a[4], 28, 32);
  if (lane == 0) {
    float lb = -lae2(v255, v256);
    if (lb > 1e29f) lb = 0.0f;                  // zero_infinity
    int tl = 0;
    for (int i = 0; i < S_; ++i) tl += (tb[i] != BLANK_) ? 1 : 0;
    float tlf = (float)((tl > 0) ? tl : 1);
    loss_ws[b] = lb / tlf;
  }
}

// Deterministic fixed-order batch mean (no float atomics).
__global__ void ctc_finalize_kernel(const float* __restrict__ loss_ws,
                                    float* __restrict__ out) {
  if (threadIdx.x == 0 && blockIdx.x == 0) {
    float s = 0.0f;
    for (int i = 0; i < B_; ++i) s += loss_ws[i];
    out[0] = s * (1.0f / (float)B_);
  }
}

extern "C" void kernel_launch(void* const* d_in, const int* in_sizes, int n_in,
                              void* d_out, int out_size, void* d_ws, size_t ws_size,
                              hipStream_t stream) {
  (void)in_sizes; (void)n_in; (void)out_size; (void)ws_size;
  const float* outputs = (const float*)d_in[0];
  const int*   targets = (const int*)d_in[1];
  float* ws = (float*)d_ws;

  ctc_alpha_kernel<<<dim3(B_), dim3(32), 0, stream>>>(outputs, targets, ws);
  ctc_finalize_kernel<<<dim3(1), dim3(1), 0, stream>>>(ws, (float*)d_out);
}